// Soft_WTConv3d_52982716563563
// MI455X (gfx1250) — compile-verified
//
#include <hip/hip_runtime.h>
#include <hip/hip_bf16.h>

// Soft_WTConv3d on MI455X (gfx1250, wave32).
//  - Haar DWT/IDWT: exact 8x8 GEMM per 2x2x2 block -> V_WMMA_F32_16X16X4_F32 x2 (K=8).
//  - Depthwise 3x3x3 convs: LDS-tiled stencils staged with the CDNA5 async
//    global->LDS path (global_load_async_to_lds_b32 / ASYNCcnt).
//  - All intermediates (<=67MB) fit in the 192MB L2; pipeline is L2-resident.

typedef __attribute__((ext_vector_type(2))) float v2f;
typedef __attribute__((ext_vector_type(8))) float v8f;

#define HAAR_C3 0.35355339059327379f  // 2^-1.5

// Decomposition Haar coefficient H[f][k]: sign = parity of bits where f=1,k=0
__device__ __forceinline__ float haar_dec(int f, int k) {
    return (__popc(f & ~k & 7) & 1) ? -HAAR_C3 : HAAR_C3;
}
// Reconstruction Haar coefficient G[f][k]: sign = parity of bits where f=1,k=1
__device__ __forceinline__ float haar_rec(int f, int k) {
    return (__popc(f & k & 7) & 1) ? -HAAR_C3 : HAAR_C3;
}

// ---------------------------------------------------------------------------
// Forward DWT: in (C, 2S, 2S, 2S) with per-channel stride in_cstride
//              out (C, 8, S, S, S) contiguous
// One wave handles 16 consecutive-w output blocks:
//   Out(16 x 8bands) = X(16 x 8) * Ht(8 x 8) via two f32 WMMAs (K=4 each)
// ---------------------------------------------------------------------------
__global__ __launch_bounds__(256) void dwt3_wmma_kernel(
    const float* __restrict__ in, long long in_cstride,
    float* __restrict__ out, int S)
{
    const int wave = blockIdx.x * (blockDim.x >> 5) + (threadIdx.x >> 5);
    const int lane = threadIdx.x & 31;

    const int wtiles = S >> 4;
    int wt = wave % wtiles;
    int tmp = wave / wtiles;
    const int h = tmp % S; tmp /= S;
    const int d = tmp % S;
    const int c = tmp / S;

    const int m    = lane & 15;   // output block (row of A, row of D)
    const int half = lane >> 4;   // lane half selects K pair {0,1} vs {2,3}
    const int kb   = half << 1;   // 0 or 2; A2/B2 use kb+4
    const int w0   = wt << 4;

    // A: X[m][k] = in[c][2d+z][2h+y][2w+x], k=(z<<2)|(y<<1)|x
    // kb pair shares z=0,y=half and is x-contiguous -> one b64 load each.
    const int S2 = 2 * S;
    const long long dst = (long long)S2 * S2;
    const long long abase = (long long)c * in_cstride
                          + (long long)(2 * d) * dst
                          + (long long)(2 * h + half) * S2
                          + (long long)(2 * (w0 + m));
    v2f a1 = *(const v2f*)(in + abase);        // z=0 (K = kb, kb+1)
    v2f a2 = *(const v2f*)(in + abase + dst);  // z=1 (K = kb+4, kb+5)

    // B[k][f] = H[f][k]; column n = lane&15, only f<8 valid (rest zeroed)
    const int n = m;
    const float mask = (n < 8) ? 1.f : 0.f;
    v2f b1, b2;
    b1.x = mask * haar_dec(n & 7, kb);
    b1.y = mask * haar_dec(n & 7, kb + 1);
    b2.x = mask * haar_dec(n & 7, kb + 4);
    b2.y = mask * haar_dec(n & 7, kb + 5);

    v8f acc = {};
    acc = __builtin_amdgcn_wmma_f32_16x16x4_f32(false, a1, false, b1,
                                                (short)0, acc, false, false);
    acc = __builtin_amdgcn_wmma_f32_16x16x4_f32(false, a2, false, b2,
                                                (short)0, acc, false, false);

    // D layout: VGPR r -> M = r (lanes 0-15) / r+8 (lanes 16-31); N = n = band f.
    if (n < 8) {
        const long long fst = (long long)S * S * S;
        const long long ob = (long long)c * 8 * fst + (long long)n * fst
                           + (long long)d * S * S + (long long)h * S
                           + (w0 + (half << 3));
        float4* op = (float4*)(out + ob);   // 16B-aligned: w0%16==0, +8 half
        op[0] = make_float4(acc[0], acc[1], acc[2], acc[3]);
        op[1] = make_float4(acc[4], acc[5], acc[6], acc[7]);
    }
}

// ---------------------------------------------------------------------------
// Inverse DWT: t (C, 8, S, S, S); optional addll (C,S,S,S) added to band 0
//              out (C, 2S, 2S, 2S) with per-channel stride out_cstride
//   Out(16 x 8elems) = X(16 x 8bands) * G(8 x 8) via two f32 WMMAs
// ---------------------------------------------------------------------------
__global__ __launch_bounds__(256) void idwt3_wmma_kernel(
    const float* __restrict__ t, const float* __restrict__ addll,
    float* __restrict__ out, long long out_cstride, int S)
{
    const int wave = blockIdx.x * (blockDim.x >> 5) + (threadIdx.x >> 5);
    const int lane = threadIdx.x & 31;

    const int wtiles = S >> 4;
    int wt = wave % wtiles;
    int tmp = wave / wtiles;
    const int h = tmp % S; tmp /= S;
    const int d = tmp % S;
    const int c = tmp / S;

    const int m    = lane & 15;
    const int half = lane >> 4;
    const int fb   = half << 1;   // A K pair: f = {fb, fb+1}, A2: {fb+4, fb+5}
    const int w0   = wt << 4;

    const long long fst = (long long)S * S * S;
    const long long sb = (long long)c * 8 * fst + (long long)d * S * S
                       + (long long)h * S + (w0 + m);
    v2f a1, a2;
    a1.x = t[sb + (long long)(fb    ) * fst];
    a1.y = t[sb + (long long)(fb + 1) * fst];
    a2.x = t[sb + (long long)(fb + 4) * fst];
    a2.y = t[sb + (long long)(fb + 5) * fst];
    if (addll != nullptr && fb == 0) {
        a1.x += addll[(long long)c * fst + (long long)d * S * S
                      + (long long)h * S + (w0 + m)];
    }

    // B[K=f][N=k] = G[f][k]
    const int n = m;
    const float mask = (n < 8) ? 1.f : 0.f;
    v2f b1, b2;
    b1.x = mask * haar_rec(fb,     n & 7);
    b1.y = mask * haar_rec(fb + 1, n & 7);
    b2.x = mask * haar_rec(fb + 4, n & 7);
    b2.y = mask * haar_rec(fb + 5, n & 7);

    v8f acc = {};
    acc = __builtin_amdgcn_wmma_f32_16x16x4_f32(false, a1, false, b1,
                                                (short)0, acc, false, false);
    acc = __builtin_amdgcn_wmma_f32_16x16x4_f32(false, a2, false, b2,
                                                (short)0, acc, false, false);

    // D: row M = block index -> w position; col n = k = (z,y,x) element.
    if (n < 8) {
        const int z = n >> 2, y = (n >> 1) & 1, x = n & 1;
        const int S2 = 2 * S;
        const long long ob = (long long)c * out_cstride
                           + (long long)(2 * d + z) * S2 * S2
                           + (long long)(2 * h + y) * S2 + x
                           + (long long)(2 * (w0 + (half << 3)));
        #pragma unroll
        for (int r = 0; r < 8; ++r) out[ob + 2 * r] = acc[r];
    }
}

// ---------------------------------------------------------------------------
// Async halo stage: copy the in-range part of a 10x10x10 halo tile for
// channel `ch` / tile origin (d0,h0,w0) into LDS via the CDNA5 async
// global->LDS path. OOB slots are never loaded and never read (zero-pad
// handled by bounds checks in the compute phase).
// ---------------------------------------------------------------------------
__device__ __forceinline__ void stage_halo_async(
    float* tile, const float* __restrict__ ip,
    int d0, int h0, int w0, int S)
{
    for (int i = threadIdx.x; i < 10 * 10 * 10; i += 256) {
        const int lw = i % 10; int t = i / 10;
        const int lh = t % 10;
        const int ld = t / 10;
        const int gd = d0 + ld, gh = h0 + lh, gw = w0 + lw;
        if ((unsigned)gd < (unsigned)S && (unsigned)gh < (unsigned)S &&
            (unsigned)gw < (unsigned)S) {
            const float* gp = ip + (long long)gd * S * S + gh * S + gw;
            // Low 32 bits of the flat shared address == LDS byte offset.
            unsigned lds_off = (unsigned)(size_t)(tile + i);
            asm volatile("global_load_async_to_lds_b32 %0, %1, off"
                         :: "v"(lds_off), "v"(gp) : "memory");
        }
    }
    asm volatile("s_wait_asynccnt 0" ::: "memory");
    __syncthreads();
}

// 3x3x3 stencil over the LDS halo tile with 'SAME' zero-pad bounds checks.
__device__ __forceinline__ float stencil27(
    const float* tile, const float* wv,
    int od, int oh, int ow, int gd, int gh, int gw, int S)
{
    float acc = 0.f;
    #pragma unroll
    for (int kz = 0; kz < 3; ++kz) {
        if ((unsigned)(gd + kz - 1) >= (unsigned)S) continue;
        #pragma unroll
        for (int ky = 0; ky < 3; ++ky) {
            if ((unsigned)(gh + ky - 1) >= (unsigned)S) continue;
            #pragma unroll
            for (int kx = 0; kx < 3; ++kx) {
                if ((unsigned)(gw + kx - 1) >= (unsigned)S) continue;
                acc = fmaf(tile[(od + kz) * 100 + (oh + ky) * 10 + (ow + kx)],
                           wv[kz * 9 + ky * 3 + kx], acc);
            }
        }
    }
    return acc;
}

// ---------------------------------------------------------------------------
// Depthwise 3x3x3 conv ('SAME') + soft-threshold of HF bands (ch%8 != 0)
// + per-channel scale. One block = one 8^3 output tile of one channel.
// ---------------------------------------------------------------------------
__global__ __launch_bounds__(256) void dwconv_soft_lds_kernel(
    const float* __restrict__ in, float* __restrict__ out,
    const float* __restrict__ wgt, const float* __restrict__ wscale,
    const float* __restrict__ lambd_p, int S)
{
    __shared__ float tile[10 * 10 * 10];

    const int tpd = S >> 3;                 // tiles per spatial dim
    int b = blockIdx.x;
    const int tw = b % tpd; b /= tpd;
    const int th = b % tpd; b /= tpd;
    const int td = b % tpd;
    const int ch = b / tpd;

    const long long csz = (long long)S * S * S;
    const float* ip = in + (long long)ch * csz;
    stage_halo_async(tile, ip, (td << 3) - 1, (th << 3) - 1, (tw << 3) - 1, S);

    const float* wp = wgt + (long long)ch * 27;
    float wv[27];
    #pragma unroll
    for (int j = 0; j < 27; ++j) wv[j] = wp[j];
    const float lam  = *lambd_p;
    const float sc   = wscale[ch];
    const bool  dohf = (ch & 7) != 0;

    for (int o = threadIdx.x; o < 512; o += 256) {
        const int ow = o & 7; int t = o >> 3;
        const int oh = t & 7;
        const int od = t >> 3;
        const int gd = (td << 3) + od, gh = (th << 3) + oh, gw = (tw << 3) + ow;
        float acc = stencil27(tile, wv, od, oh, ow, gd, gh, gw, S);
        if (dohf) {  // HF band: sign(t) * relu(|t| - lambda)
            const float mag = fabsf(acc) - lam;
            acc = (mag > 0.f) ? copysignf(mag, acc) : 0.f;
        }
        out[(long long)ch * csz + (long long)gd * S * S + gh * S + gw] = acc * sc;
    }
}

// ---------------------------------------------------------------------------
// base = (depthwise3x3x3(x, base_w) + base_b) * base_scale ; out = base + nxt
// Same LDS-tiled structure, S=64.
// ---------------------------------------------------------------------------
__global__ __launch_bounds__(256) void base_add_lds_kernel(
    const float* __restrict__ x, const float* __restrict__ bw,
    const float* __restrict__ bb, const float* __restrict__ bscale,
    const float* __restrict__ nxt, float* __restrict__ out, int S)
{
    __shared__ float tile[10 * 10 * 10];

    const int tpd = S >> 3;
    int b = blockIdx.x;
    const int tw = b % tpd; b /= tpd;
    const int th = b % tpd; b /= tpd;
    const int td = b % tpd;
    const int ch = b / tpd;

    const long long csz = (long long)S * S * S;
    const float* ip = x + (long long)ch * csz;
    stage_halo_async(tile, ip, (td << 3) - 1, (th << 3) - 1, (tw << 3) - 1, S);

    const float* wp = bw + (long long)ch * 27;
    float wv[27];
    #pragma unroll
    for (int j = 0; j < 27; ++j) wv[j] = wp[j];
    const float bias = bb[ch];
    const float sc   = bscale[ch];

    for (int o = threadIdx.x; o < 512; o += 256) {
        const int ow = o & 7; int t = o >> 3;
        const int oh = t & 7;
        const int od = t >> 3;
        const int gd = (td << 3) + od, gh = (th << 3) + oh, gw = (tw << 3) + ow;
        const float acc = stencil27(tile, wv, od, oh, ow, gd, gh, gw, S);
        const long long idx = (long long)ch * csz + (long long)gd * S * S
                            + gh * S + gw;
        out[idx] = (acc + bias) * sc + nxt[idx];
    }
}

// ---------------------------------------------------------------------------
extern "C" void kernel_launch(void* const* d_in, const int* in_sizes, int n_in,
                              void* d_out, int out_size, void* d_ws, size_t ws_size,
                              hipStream_t stream) {
    const float* x          = (const float*)d_in[0]; // (1,64,64,64,64)
    const float* base_w     = (const float*)d_in[1]; // (64,1,3,3,3)
    const float* base_b     = (const float*)d_in[2]; // (64,)
    const float* base_scale = (const float*)d_in[3]; // (1,64,1,1,1)
    const float* wconv_w0   = (const float*)d_in[4]; // (512,1,3,3,3)
    const float* wconv_w1   = (const float*)d_in[5]; // (512,1,3,3,3)
    const float* wscale0    = (const float*)d_in[6]; // (1,512,1,1,1)
    const float* wscale1    = (const float*)d_in[7]; // (1,512,1,1,1)
    const float* lambd      = (const float*)d_in[8]; // scalar

    const int C = 64;
    float* ws = (float*)d_ws;
    // Workspace layout (floats). NXT reuses CX0: CX0 is dead after dwt1.
    float* CX0  = ws;                       // (64,8,32^3)  16,777,216
    float* T0   = CX0 + 16777216LL;         // (64,8,32^3)  16,777,216
    float* CX1  = T0  + 16777216LL;         // (64,8,16^3)   2,097,152
    float* T1   = CX1 + 2097152LL;          // (64,8,16^3)   2,097,152
    float* REC1 = T1  + 2097152LL;          // (64,32^3)     2,097,152
    float* NXT  = CX0;                      // (64,64^3)    16,777,216

    // 1. Level-0 DWT: x (c-stride 64^3) -> CX0, S_out=32
    {
        const int tiles = C * 32 * 32 * 2;          // (c,d,h,wtile)
        dwt3_wmma_kernel<<<tiles / 8, 256, 0, stream>>>(x, 262144LL, CX0, 32);
    }
    // 2. Level-0 depthwise conv + soft-threshold + scale: CX0 -> T0
    {
        const int blocks = 512 * 4 * 4 * 4;         // (ch, 8^3-tiles over 32^3)
        dwconv_soft_lds_kernel<<<blocks, 256, 0, stream>>>(
            CX0, T0, wconv_w0, wscale0, lambd, 32);
    }
    // 3. Level-1 DWT on raw LL band of CX0 (f=0, c-stride 8*32^3) -> CX1, S_out=16
    {
        const int tiles = C * 16 * 16 * 1;
        dwt3_wmma_kernel<<<tiles / 8, 256, 0, stream>>>(CX0, 262144LL, CX1, 16);
    }
    // 4. Level-1 depthwise conv + soft-threshold + scale: CX1 -> T1
    {
        const int blocks = 512 * 2 * 2 * 2;         // (ch, 8^3-tiles over 16^3)
        dwconv_soft_lds_kernel<<<blocks, 256, 0, stream>>>(
            CX1, T1, wconv_w1, wscale1, lambd, 16);
    }
    // 5. Level-1 IDWT: T1 -> REC1 (c-stride 32^3)
    {
        const int tiles = C * 16 * 16 * 1;
        idwt3_wmma_kernel<<<tiles / 8, 256, 0, stream>>>(T1, nullptr, REC1, 32768LL, 16);
    }
    // 6. Level-0 IDWT with LL += REC1: T0 -> NXT (c-stride 64^3)
    {
        const int tiles = C * 32 * 32 * 2;
        idwt3_wmma_kernel<<<tiles / 8, 256, 0, stream>>>(T0, REC1, NXT, 262144LL, 32);
    }
    // 7. Base depthwise conv + bias + scale + NXT -> d_out
    {
        const int blocks = 64 * 8 * 8 * 8;          // (c, 8^3-tiles over 64^3)
        base_add_lds_kernel<<<blocks, 256, 0, stream>>>(
            x, base_w, base_b, base_scale, NXT, (float*)d_out, 64);
    }
}